// HybridAblationModel_49718541418747
// MI455X (gfx1250) — compile-verified
//
#include <hip/hip_runtime.h>

typedef __attribute__((ext_vector_type(2))) float v2f;
typedef __attribute__((ext_vector_type(8))) float v8f;

// ---------------- compile-time Cl(4,1) tables ----------------
struct Tables {
  float sgn[32][32];   // sgn[i][k] = C[i, i^k, k]  (e_i * e_{i^k} -> e_k coefficient)
  float qs[32];        // reversion * metric sign per blade
};
constexpr int pc_(int x){ int c=0; while(x){ c += x & 1; x >>= 1; } return c; }
constexpr Tables make_tables(){
  Tables t{};
  for (int i = 0; i < 32; ++i){
    for (int k = 0; k < 32; ++k){
      const int b = i ^ k;
      int s = 0, ash = i >> 1;
      while (ash){ s += pc_(ash & b); ash >>= 1; }
      float g = (s & 1) ? -1.f : 1.f;
      if ((i & b) & 16) g = -g;          // metric = [1,1,1,1,-1]
      t.sgn[i][k] = g;
    }
  }
  for (int k = 0; k < 32; ++k){
    const int r = pc_(k);
    const float rev = ((r*(r-1)/2) & 1) ? -1.f : 1.f;
    t.qs[k] = (k & 16) ? -rev : rev;
  }
  return t;
}
constexpr Tables TAB = make_tables();

// U staging layout: row (bn) stride 580 floats (bank-skew 4), per-h stride 36 floats
#define U_BN_STRIDE 580
#define U_H_STRIDE  36

__global__ void __launch_bounds__(256)
clifford_scan_kernel(const float* __restrict__ x,
                     const float* __restrict__ W_in,
                     const float* __restrict__ b_in,
                     const float* __restrict__ W_out,
                     const float* __restrict__ b_out,
                     float* __restrict__ out)
{
  __shared__ float lds_Win [6 * 512];          // 12 KB  W_in[d][col]
  __shared__ float lds_Wout[512 * 6];          // 12 KB  W_out[k][d]
  __shared__ float lds_U   [16 * U_BN_STRIDE]; // 36.25 KB  u / psi staging
  __shared__ float lds_X   [16 * 8];           // x_t rows, D padded 6->8 (pad = 0)
  __shared__ float lds_PRED[16 * 16];          // output tile (cols 6..15 = scratch)

  const int tid  = threadIdx.x;
  const int lane = tid & 31;
  const int wave = tid >> 5;
  const int bnl  = tid >> 4;        // local (b,n) row 0..15
  const int h    = tid & 15;        // channel within (b,n)
  const int mlo  = lane & 15;
  const int mloC = (mlo < 6) ? mlo : 5;   // clamped W_out column (cols >=6 never read)
  const int halfK = (lane < 16) ? 0 : 2;
  const int mhi   = (lane < 16) ? 0 : 8;

  const int gb = blockIdx.x * 16;   // global flat bn base (block owns 16 rows)
  const int bb = gb >> 9;           // batch index (N = 512)
  const int n0 = gb & 511;
  const int xbase0 = ((bb * 64) * 512 + n0) * 6;

  // --------- one-time init ---------
  for (int i = tid; i < 6 * 512; i += 256) lds_Win[i]  = W_in[i];
  for (int i = tid; i < 512 * 6; i += 256) lds_Wout[i] = W_out[i];
  if (tid < 16){ lds_X[tid*8 + 6] = 0.f; lds_X[tid*8 + 7] = 0.f; }

  float bin[32];
  #pragma unroll
  for (int q = 0; q < 8; ++q){
    float4 t4 = *reinterpret_cast<const float4*>(&b_in[h*32 + q*4]);
    bin[q*4+0] = t4.x; bin[q*4+1] = t4.y; bin[q*4+2] = t4.z; bin[q*4+3] = t4.w;
  }
  float bo = 0.f;
  if (tid < 96) bo = b_out[tid % 6];

  float psi[32];
  #pragma unroll
  for (int k = 0; k < 32; ++k) psi[k] = 0.f;
  psi[0] = 1.f;

  const int ubase = bnl * U_BN_STRIDE + h * U_H_STRIDE;

  __syncthreads();

  for (int t = 0; t < 64; ++t){
    __syncthreads();                                   // (A) prev-step consumers done
    // S0: x_t -> LDS ; S1: zero PRED
    if (tid < 96){
      const int r = tid / 6, d = tid - r*6;
      lds_X[r*8 + d] = x[xbase0 + t*3072 + tid];
    }
    lds_PRED[tid] = 0.f;
    __syncthreads();                                   // (B) X ready

    // ---- input projection: U[16,512] = X[16,6(pad8)] @ W_in, wmma f32 16x16x4 ----
    // A rows are zero for k>=6, so B can read clamped (valid) addresses unguarded.
    for (int nt = wave*4; nt < wave*4 + 4; ++nt){
      v8f acc = {};
      const int col = nt*16 + mlo;
      #pragma unroll
      for (int kb = 0; kb < 8; kb += 4){
        const int k0 = kb + halfK, k1 = k0 + 1;
        const int k0c = (k0 < 6) ? k0 : 5;
        const int k1c = (k1 < 6) ? k1 : 5;
        v2f a, bfr;
        a.x = lds_X[mlo*8 + k0];
        a.y = lds_X[mlo*8 + k1];
        bfr.x = lds_Win[k0c*512 + col];
        bfr.y = lds_Win[k1c*512 + col];
        acc = __builtin_amdgcn_wmma_f32_16x16x4_f32(false, a, false, bfr,
                                                    (short)0, acc, false, false);
      }
      const int colOff = (col >> 5) * U_H_STRIDE + (col & 31);
      #pragma unroll
      for (int r = 0; r < 8; ++r)
        lds_U[(r + mhi) * U_BN_STRIDE + colOff] = acc[r];
    }
    __syncthreads();                                   // (C) U ready

    // ---- per-channel delta = normalize(u + b_in + e0) ----
    float delta[32];
    #pragma unroll
    for (int q = 0; q < 16; ++q){
      const float2 uv = *reinterpret_cast<const float2*>(&lds_U[ubase + q*2]);
      delta[q*2]   = uv.x + bin[q*2];
      delta[q*2+1] = uv.y + bin[q*2+1];
    }
    delta[0] += 1.0f;
    {
      float qv = 0.f;
      #pragma unroll
      for (int k = 0; k < 32; ++k) qv = fmaf(TAB.qs[k] * delta[k], delta[k], qv);
      const float rn = rsqrtf(fmaxf(fabsf(qv), 1e-6f));
      #pragma unroll
      for (int k = 0; k < 32; ++k) delta[k] *= rn;
    }

    // ---- geometric product (1024 FMAs, signs folded to modifiers) + normalize ----
    float accp[32];
    #pragma unroll
    for (int k = 0; k < 32; ++k) accp[k] = 0.f;
    #pragma unroll
    for (int i = 0; i < 32; ++i){
      #pragma unroll
      for (int k = 0; k < 32; ++k)
        accp[k] = fmaf(TAB.sgn[i][k] * delta[i], psi[i ^ k], accp[k]);
    }
    {
      float qv = 0.f;
      #pragma unroll
      for (int k = 0; k < 32; ++k) qv = fmaf(TAB.qs[k] * accp[k], accp[k], qv);
      const float rn = rsqrtf(fmaxf(fabsf(qv), 1e-6f));
      #pragma unroll
      for (int k = 0; k < 32; ++k) psi[k] = accp[k] * rn;
    }
    __syncthreads();                                   // (D) all U reads done

    // ---- stage psi for output projection ----
    #pragma unroll
    for (int q = 0; q < 16; ++q){
      float2 pv; pv.x = psi[q*2]; pv.y = psi[q*2+1];
      *reinterpret_cast<float2*>(&lds_U[ubase + q*2]) = pv;
    }
    __syncthreads();                                   // (E) psi staged

    // ---- output projection: PRED[16,6] = PSI[16,512] @ W_out; K split over 8 waves ----
    // B columns >=6 read clamped col 5; resulting PRED cols 6..15 are never read.
    {
      v8f acc = {};
      for (int kt = wave*16; kt < wave*16 + 16; ++kt){
        const int k0 = kt*4 + halfK, k1 = k0 + 1;
        v2f a, bfr;
        a.x = lds_U[mlo * U_BN_STRIDE + (k0 >> 5) * U_H_STRIDE + (k0 & 31)];
        a.y = lds_U[mlo * U_BN_STRIDE + (k1 >> 5) * U_H_STRIDE + (k1 & 31)];
        bfr.x = lds_Wout[k0*6 + mloC];
        bfr.y = lds_Wout[k1*6 + mloC];
        acc = __builtin_amdgcn_wmma_f32_16x16x4_f32(false, a, false, bfr,
                                                    (short)0, acc, false, false);
      }
      #pragma unroll
      for (int r = 0; r < 8; ++r)
        atomicAdd(&lds_PRED[(r + mhi)*16 + mlo], acc[r]);
    }
    __syncthreads();                                   // (F) PRED complete

    // ---- out = x_t + pred + b_out ----
    if (tid < 96){
      const int r = tid / 6, d = tid - r*6;
      out[xbase0 + t*3072 + tid] = lds_X[r*8 + d] + lds_PRED[r*16 + d] + bo;
    }
  }
}

extern "C" void kernel_launch(void* const* d_in, const int* in_sizes, int n_in,
                              void* d_out, int out_size, void* d_ws, size_t ws_size,
                              hipStream_t stream) {
  (void)in_sizes; (void)n_in; (void)out_size; (void)d_ws; (void)ws_size;
  const float* x     = (const float*)d_in[0];
  const float* W_in  = (const float*)d_in[1];
  const float* b_in  = (const float*)d_in[2];
  const float* W_out = (const float*)d_in[3];
  const float* b_out = (const float*)d_in[4];
  float* out = (float*)d_out;
  // 2048 (b,n) rows / 16 rows per block = 128 blocks, 256 threads (8 wave32)
  clifford_scan_kernel<<<dim3(128), dim3(256), 0, stream>>>(x, W_in, b_in, W_out, b_out, out);
}